// CELoss_15745350107749
// MI455X (gfx1250) — compile-verified
//
#include <hip/hip_runtime.h>
#include <stdint.h>

// ---------------------------------------------------------------------------
// ECE/MCE calibration-error pipeline for MI455X (gfx1250, wave32).
// Bandwidth-bound: 1.07 GB of logits @ 23.3 TB/s ~ 46 us floor.
// Big pass streams rows through LDS with CDNA5 async global->LDS loads
// (ASYNCcnt pipeline), row-double-buffered so the async engine never drains.
// ---------------------------------------------------------------------------

#define NBINS     20
#define ROW_C     1024              // columns (fixed by problem)
#define CHUNKS    8                 // 8 x 128 floats = 1024
#define HBUCKETS  262144            // covers key = float_bits(conf) >> 12 (conf <= 1.0)

// issue 8 async B128 loads: per lane 16B, chunk stride 512B; the 24-bit
// instruction offset is added to BOTH the global and the LDS address, so one
// (lds, global) base pair per lane serves the whole 4 KB row.
// th:TH_LOAD_NT: logits are single-use -> don't rinse the 192 MB L2.
#define ISSUE_ROW_ASYNC(LDSA, GA)                                             \
  asm volatile(                                                               \
      "global_load_async_to_lds_b128 %0, %1, off th:TH_LOAD_NT\n"             \
      "global_load_async_to_lds_b128 %0, %1, off offset:512 th:TH_LOAD_NT\n"  \
      "global_load_async_to_lds_b128 %0, %1, off offset:1024 th:TH_LOAD_NT\n" \
      "global_load_async_to_lds_b128 %0, %1, off offset:1536 th:TH_LOAD_NT\n" \
      "global_load_async_to_lds_b128 %0, %1, off offset:2048 th:TH_LOAD_NT\n" \
      "global_load_async_to_lds_b128 %0, %1, off offset:2560 th:TH_LOAD_NT\n" \
      "global_load_async_to_lds_b128 %0, %1, off offset:3072 th:TH_LOAD_NT\n" \
      "global_load_async_to_lds_b128 %0, %1, off offset:3584 th:TH_LOAD_NT\n" \
      :: "v"(LDSA), "v"(GA) : "memory")

// ---- Kernel 1: per-row softmax-max / argmax / sum-exp via async LDS stream --

__global__ __launch_bounds__(256) void ece_rowstat_kernel(
    const float* __restrict__ logits, const int* __restrict__ labels,
    float* __restrict__ conf, unsigned* __restrict__ acc,
    unsigned* __restrict__ hist, int n) {
  // 8 waves/block, 2 row buffers of 4 KB per wave -> 64 KB/block
  __shared__ float4 sbuf[8 * 2 * 256];

  const int lane = threadIdx.x & 31;
  const int wid  = threadIdx.x >> 5;
  const int gw   = (blockIdx.x * blockDim.x + threadIdx.x) >> 5;
  const int nw   = (gridDim.x * blockDim.x) >> 5;

  const float4* vbuf0 = &sbuf[wid * 512];
  const float4* vbuf1 = vbuf0 + 256;
  // low 32 bits of a flat shared-aperture pointer == workgroup-relative LDS address
  const unsigned lds0 = (unsigned)(uintptr_t)(const void*)(vbuf0 + lane);
  const unsigned lds1 = lds0 + 4096;

  int r = gw;
  if (r < n) {  // prologue: prefetch first row into buffer 0
    unsigned long long ga =
        (unsigned long long)(uintptr_t)(logits + (size_t)r * ROW_C + (lane << 2));
    ISSUE_ROW_ASYNC(lds0, ga);
  }

  int cur = 0;
  for (; r < n; r += nw) {
    const int nxt = r + nw;
    if (nxt < n) {
      // prefetch next row into the other buffer, then wait until only those
      // 8 ops are outstanding: async loads complete in order, so the current
      // row's 8 transfers are guaranteed done.
      unsigned ldsn = cur ? lds0 : lds1;
      unsigned long long gan =
          (unsigned long long)(uintptr_t)(logits + (size_t)nxt * ROW_C + (lane << 2));
      ISSUE_ROW_ASYNC(ldsn, gan);
      asm volatile("s_wait_asynccnt 8" ::: "memory");
    } else {
      asm volatile("s_wait_asynccnt 0" ::: "memory");
    }
    const float4* vb = cur ? vbuf1 : vbuf0;

    // ---- pass 1: row max + first-occurrence argmax ----
    float m = -3.0e38f;
    int idx = 0;
    #pragma unroll
    for (int k = 0; k < CHUNKS; ++k) {
      float4 v = vb[k * 32 + lane];
      int base = k * 128 + (lane << 2);
      if (v.x > m) { m = v.x; idx = base;     }
      if (v.y > m) { m = v.y; idx = base + 1; }
      if (v.z > m) { m = v.z; idx = base + 2; }
      if (v.w > m) { m = v.w; idx = base + 3; }
    }
    // wave32 butterfly: max with smaller-index tiebreak (first occurrence)
    #pragma unroll
    for (int off = 16; off > 0; off >>= 1) {
      float om = __shfl_xor(m, off, 32);
      int   oi = __shfl_xor(idx, off, 32);
      if (om > m || (om == m && oi < idx)) { m = om; idx = oi; }
    }

    // ---- pass 2: sum exp(x - M) from LDS (one v_exp_f32 per element) ----
    float s = 0.0f;
    #pragma unroll
    for (int k = 0; k < CHUNKS; ++k) {
      float4 v = vb[k * 32 + lane];
      s += __expf(v.x - m);
      s += __expf(v.y - m);
      s += __expf(v.z - m);
      s += __expf(v.w - m);
    }
    #pragma unroll
    for (int off = 16; off > 0; off >>= 1) s += __shfl_xor(s, off, 32);

    if (lane == 0) {
      float cf = 1.0f / s;                 // max softmax = exp(M-M)/sum = 1/s
      int lab = labels[r];
      conf[r] = cf;
      acc[r]  = (idx == lab) ? 1u : 0u;
      unsigned key = __float_as_uint(cf) >> 12;   // positive float: monotone
      atomicAdd(&hist[key], 1u);
    }
    cur ^= 1;
  }
}

// ---- Kernel 2: exclusive prefix sum of the bucket histogram ---------------

__global__ void ece_scan_kernel(const unsigned* __restrict__ hist,
                                unsigned* __restrict__ cursor, int hb) {
  __shared__ unsigned tmp[1024];
  const int t = threadIdx.x;
  unsigned carry = 0;
  for (int base = 0; base < hb; base += 1024) {
    unsigned v = hist[base + t];
    tmp[t] = v;
    __syncthreads();
    #pragma unroll
    for (int off = 1; off < 1024; off <<= 1) {
      unsigned y = (t >= off) ? tmp[t - off] : 0u;
      __syncthreads();
      tmp[t] += y;
      __syncthreads();
    }
    cursor[base + t] = carry + tmp[t] - v;   // exclusive
    carry += tmp[1023];
    __syncthreads();
  }
}

// ---- Kernel 3: rank -> bin, fixed-point per-bin sums ----------------------

__global__ __launch_bounds__(256) void ece_binsum_kernel(
    const float* __restrict__ conf, const unsigned* __restrict__ acc,
    unsigned* __restrict__ cursor,
    unsigned long long* __restrict__ binConf,
    unsigned long long* __restrict__ binAcc, int n, int bin_size) {
  __shared__ unsigned long long sc[NBINS];
  __shared__ unsigned long long sa[NBINS];
  if (threadIdx.x < NBINS) { sc[threadIdx.x] = 0ull; sa[threadIdx.x] = 0ull; }
  __syncthreads();

  const int stride = gridDim.x * blockDim.x;
  for (int i = blockIdx.x * blockDim.x + threadIdx.x; i < n; i += stride) {
    float c = conf[i];
    unsigned b = __float_as_uint(c) >> 12;
    unsigned rank = atomicAdd(&cursor[b], 1u);       // bucket base + ordinal
    unsigned bin = rank / (unsigned)bin_size;
    if (bin >= NBINS) bin = NBINS - 1;
    // fixed-point Q2.30 keeps the sum order-independent
    unsigned long long fx = (unsigned long long)(c * 1073741824.0f + 0.5f);
    atomicAdd(&sc[bin], fx);
    atomicAdd(&sa[bin], (unsigned long long)acc[i]);
  }
  __syncthreads();
  if (threadIdx.x < NBINS) {
    atomicAdd(&binConf[threadIdx.x], sc[threadIdx.x]);
    atomicAdd(&binAcc[threadIdx.x], sa[threadIdx.x]);
  }
}

// ---- Kernel 4: ce -> ece, mce ---------------------------------------------

__global__ void ece_finalize_kernel(const unsigned long long* __restrict__ binConf,
                                    const unsigned long long* __restrict__ binAcc,
                                    float* __restrict__ out, int bin_size) {
  if (threadIdx.x == 0 && blockIdx.x == 0) {
    float ece = 0.0f, mce = 0.0f;
    for (int b = 0; b < NBINS; ++b) {
      double scf = (double)binConf[b] * (1.0 / 1073741824.0);
      double saf = (double)binAcc[b];
      float ce = (float)(fabs(scf - saf) / (double)bin_size);
      ece += ce;
      mce = fmaxf(mce, ce);
    }
    out[0] = ece / (float)NBINS;
    out[1] = mce;
  }
}

// ---------------------------------------------------------------------------

extern "C" void kernel_launch(void* const* d_in, const int* in_sizes, int n_in,
                              void* d_out, int out_size, void* d_ws, size_t ws_size,
                              hipStream_t stream) {
  const float* logits = (const float*)d_in[0];
  const int*   labels = (const int*)d_in[1];
  const int n = in_sizes[1];           // 260000 rows
  const int bin_size = n / NBINS;      // 13000

  size_t off = 0;
  auto carve = [&](size_t bytes) {
    void* p = (char*)d_ws + off;
    off += (bytes + 255) & ~(size_t)255;
    return p;
  };
  float*              conf   = (float*)carve((size_t)n * sizeof(float));
  unsigned*           acc    = (unsigned*)carve((size_t)n * sizeof(unsigned));
  unsigned*           hist   = (unsigned*)carve((size_t)HBUCKETS * sizeof(unsigned));
  unsigned*           cursor = (unsigned*)carve((size_t)HBUCKETS * sizeof(unsigned));
  unsigned long long* bins   = (unsigned long long*)carve(2 * NBINS * sizeof(unsigned long long));
  if (off > ws_size) return;  // insufficient scratch

  hipMemsetAsync(hist, 0, (size_t)HBUCKETS * sizeof(unsigned), stream);
  hipMemsetAsync(bins, 0, 2 * NBINS * sizeof(unsigned long long), stream);

  ece_rowstat_kernel<<<2048, 256, 0, stream>>>(logits, labels, conf, acc, hist, n);
  ece_scan_kernel<<<1, 1024, 0, stream>>>(hist, cursor, HBUCKETS);
  ece_binsum_kernel<<<1024, 256, 0, stream>>>(conf, acc, cursor, bins, bins + NBINS,
                                              n, bin_size);
  ece_finalize_kernel<<<1, 32, 0, stream>>>(bins, bins + NBINS, (float*)d_out, bin_size);
}